// RENRG_27865747816667
// MI455X (gfx1250) — compile-verified
//
#include <hip/hip_runtime.h>
#include <hip/hip_bf16.h>

// ---------------------------------------------------------------------------
// Problem dimensions (match reference)
// ---------------------------------------------------------------------------
enum { N_ = 512, M_ = 512, NXI_ = 1024, L_ = 1024, S_ = 512, D3_ = 3072 };
#define EPS_ 0.001f

typedef __attribute__((ext_vector_type(2))) float v2f;
typedef __attribute__((ext_vector_type(8))) float v8f;

// ---------------------------------------------------------------------------
// Generic f32 WMMA GEMM:  C = alpha_c * g^gpow * op(A) @ op(B) + beta * C
// Block tile 128x128, K staged in LDS in chunks of 16 (= 4 WMMA k-steps of 4).
// 256 threads = 8 wave32 waves arranged 2x4; each wave owns a 64x32 sub-tile
// (4x2 grid of 16x16 WMMA tiles) so every A fragment feeds 2 WMMAs and every
// B fragment feeds 4. Global->LDS staging is float4 (global_load_b128) along
// whichever global dimension is contiguous. All dims used are multiples of
// 128 (M,N) and 16 (K) -> no edge handling.
// ---------------------------------------------------------------------------
#define BM 128
#define BN 128
#define BK 16
#define LDP (BM + 4)   // padded LDS row stride (floats), keeps 16B alignment

__device__ __forceinline__ const float* a_tile_ptr(const float* A, int lda,
                                                   int transA, int kt, int bm,
                                                   int v)
{
    if (transA) {               // op(A)[m][k] = A[k][m]; m contiguous
        int k = v >> 5, m4 = (v & 31) * 4;
        return &A[(size_t)(kt + k) * lda + bm + m4];
    } else {                    // k contiguous
        int m = v >> 2, k4 = (v & 3) * 4;
        return &A[(size_t)(bm + m) * lda + kt + k4];
    }
}

__device__ __forceinline__ const float* b_tile_ptr(const float* B, int ldb,
                                                   int transB, int kt, int bn,
                                                   int v)
{
    if (transB) {               // op(B)[k][n] = B[n][k]; k contiguous
        int n = v >> 2, k4 = (v & 3) * 4;
        return &B[(size_t)(bn + n) * ldb + kt + k4];
    } else {                    // n contiguous
        int k = v >> 5, n4 = (v & 31) * 4;
        return &B[(size_t)(kt + k) * ldb + bn + n4];
    }
}

__global__ __launch_bounds__(256) void wmma_gemm_f32(
    const float* __restrict__ A, const float* __restrict__ B,
    float* __restrict__ C,
    int Mdim, int Ndim, int Kdim, int lda, int ldb, int ldc,
    int transA, int transB,
    float alpha_c, const float* __restrict__ gptr, int gpow, float beta)
{
    __shared__ float As[BK][LDP];
    __shared__ float Bs[BK][LDP];

    const int tid   = threadIdx.x;
    const int lane  = tid & 31;          // wave32
    const int wave  = tid >> 5;          // 0..7
    const int wm    = (wave & 1) * 64;   // wave sub-tile row offset
    const int wn    = (wave >> 1) * 32;  // wave sub-tile col offset
    const int bm    = blockIdx.y * BM;
    const int bn    = blockIdx.x * BN;
    const int lhalf = lane >> 4;         // 0: k-pair {0,1}, 1: k-pair {2,3}
    const int lmod  = lane & 15;

    v8f acc[4][2];
#pragma unroll
    for (int i = 0; i < 4; ++i)
#pragma unroll
        for (int j = 0; j < 2; ++j) {
            v8f z = {0.f, 0.f, 0.f, 0.f, 0.f, 0.f, 0.f, 0.f};
            acc[i][j] = z;
        }

    for (int kt = 0; kt < Kdim; kt += BK) {
        // Prefetch next K tile toward this WGP (global_prefetch_b8); entire
        // working set is L2-resident so this mostly hides L2->WGP latency.
        if (kt + BK < Kdim) {
            __builtin_prefetch(a_tile_ptr(A, lda, transA, kt + BK, bm, tid), 0, 1);
            __builtin_prefetch(b_tile_ptr(B, ldb, transB, kt + BK, bn, tid), 0, 1);
        }

        // Stage A tile As[k][m] and B tile Bs[k][n]: 16x128 floats each =
        // 512 float4 loads per matrix, 2 per thread, all global_load_b128.
#pragma unroll
        for (int r = 0; r < 2; ++r) {
            int v = tid + r * 256;
            float4 va = *(const float4*)a_tile_ptr(A, lda, transA, kt, bm, v);
            if (transA) {
                int k = v >> 5, m4 = (v & 31) * 4;
                *(float4*)&As[k][m4] = va;
            } else {
                int m = v >> 2, k4 = (v & 3) * 4;
                As[k4 + 0][m] = va.x;
                As[k4 + 1][m] = va.y;
                As[k4 + 2][m] = va.z;
                As[k4 + 3][m] = va.w;
            }
            float4 vb = *(const float4*)b_tile_ptr(B, ldb, transB, kt, bn, v);
            if (transB) {
                int n = v >> 2, k4 = (v & 3) * 4;
                Bs[k4 + 0][n] = vb.x;
                Bs[k4 + 1][n] = vb.y;
                Bs[k4 + 2][n] = vb.z;
                Bs[k4 + 3][n] = vb.w;
            } else {
                int k = v >> 5, n4 = (v & 31) * 4;
                *(float4*)&Bs[k][n4] = vb;
            }
        }
        __syncthreads();

#pragma unroll
        for (int kk = 0; kk < BK; kk += 4) {
            // A frag (16x4, ISA layout): lane = m + 16*(k>>1), reg = k&1
            v2f afrag[4], bfrag[2];
#pragma unroll
            for (int ti = 0; ti < 4; ++ti) {
                afrag[ti].x = As[kk + lhalf * 2 + 0][wm + ti * 16 + lmod];
                afrag[ti].y = As[kk + lhalf * 2 + 1][wm + ti * 16 + lmod];
            }
#pragma unroll
            for (int tj = 0; tj < 2; ++tj) {
                bfrag[tj].x = Bs[kk + lhalf * 2 + 0][wn + tj * 16 + lmod];
                bfrag[tj].y = Bs[kk + lhalf * 2 + 1][wn + tj * 16 + lmod];
            }
#pragma unroll
            for (int ti = 0; ti < 4; ++ti)
#pragma unroll
                for (int tj = 0; tj < 2; ++tj)
                    acc[ti][tj] = __builtin_amdgcn_wmma_f32_16x16x4_f32(
                        false, afrag[ti], false, bfrag[tj],
                        (short)0, acc[ti][tj], false, false);
        }
        __syncthreads();
    }

    float alpha = alpha_c;
    if (gpow > 0)      alpha *= gptr[0];
    else if (gpow < 0) alpha /= gptr[0];

    // C/D layout: lanes 0-15: n=lane, m=reg; lanes 16-31: n=lane-16, m=reg+8
#pragma unroll
    for (int ti = 0; ti < 4; ++ti)
#pragma unroll
        for (int tj = 0; tj < 2; ++tj)
#pragma unroll
            for (int r = 0; r < 8; ++r) {
                int m = bm + wm + ti * 16 + lhalf * 8 + r;
                int n = bn + wn + tj * 16 + lmod;
                size_t off = (size_t)m * ldc + n;
                float out = alpha * acc[ti][tj][r];
                if (beta != 0.0f) out += beta * C[off];
                C[off] = out;
            }
}

// ---------------------------------------------------------------------------
// Single-workgroup Gauss-Jordan (no pivoting). aug is n x ncols row-major.
// Inverse: ncols = 2n (right half holds A^-1). Solve: ncols = n+1.
// ---------------------------------------------------------------------------
__global__ __launch_bounds__(1024) void gauss_jordan(float* __restrict__ aug,
                                                     int n, int ncols)
{
    __shared__ float prow[2080];
    __shared__ float fcol[1024];
    __shared__ float s_piv;
    const int tid = threadIdx.x;
    const int bs  = blockDim.x;

    for (int k = 0; k < n; ++k) {
        if (tid == 0) s_piv = aug[(size_t)k * ncols + k];
        __syncthreads();
        float pinv = 1.0f / s_piv;
        for (int j = tid; j < ncols; j += bs) {
            float v = aug[(size_t)k * ncols + j] * pinv;
            aug[(size_t)k * ncols + j] = v;
            prow[j] = v;
        }
        for (int i = tid; i < n; i += bs)
            fcol[i] = aug[(size_t)i * ncols + k];
        __syncthreads();
        for (int i = 0; i < n; ++i) {
            if (i == k) continue;
            float f = fcol[i];
            for (int j = tid; j < ncols; j += bs)
                aug[(size_t)i * ncols + j] =
                    fmaf(-f, prow[j], aug[(size_t)i * ncols + j]);
        }
        __syncthreads();
    }
}

// ---------------------------------------------------------------------------
// Sequential ReLU forward-substitution: eps[i] = relu((drive[i] - sum_{j<i}
// H22[i][j]*eps[j]) / H22[i][i])   (D11 = -tril(H22,-1), Lam = diag(H22))
// ---------------------------------------------------------------------------
__global__ __launch_bounds__(1024) void seq_eps_kernel(
    const float* __restrict__ H, const float* __restrict__ drive,
    float* __restrict__ eps)
{
    __shared__ float s_e;
    const int tid = threadIdx.x;
    float acc = drive[tid];
    float lam = H[(size_t)(NXI_ + tid) * D3_ + NXI_ + tid];

    for (int i = 0; i < L_; ++i) {
        if (tid == i) {
            float e = acc / lam;
            e = e > 0.0f ? e : 0.0f;
            s_e = e;
            eps[i] = e;
        }
        __syncthreads();
        float e = s_e;
        if (tid > i)
            acc -= H[(size_t)(NXI_ + tid) * D3_ + NXI_ + i] * e;
        __syncthreads();
    }
}

// ---------------------------------------------------------------------------
// Block-per-row reduction helpers + matvec kernels
// ---------------------------------------------------------------------------
__device__ __forceinline__ float block_dot(const float* __restrict__ a,
                                           const float* __restrict__ b,
                                           int n, float* red)
{
    float s = 0.0f;
    for (int j = threadIdx.x; j < n; j += blockDim.x) s = fmaf(a[j], b[j], s);
    __syncthreads();
    red[threadIdx.x] = s;
    __syncthreads();
    for (int w = blockDim.x >> 1; w > 0; w >>= 1) {
        if ((int)threadIdx.x < w) red[threadIdx.x] += red[threadIdx.x + w];
        __syncthreads();
    }
    return red[0];
}

// drive[r] = -(H21[r,:] @ xi) + D12[r,:] @ w
__global__ __launch_bounds__(256) void drive_kernel(
    const float* __restrict__ H, const float* __restrict__ D12,
    const float* __restrict__ xi, const float* __restrict__ w,
    float* __restrict__ drive)
{
    __shared__ float red[256];
    const int r = blockIdx.x;
    float d1 = block_dot(H + (size_t)(NXI_ + r) * D3_, xi, NXI_, red);
    float d2 = block_dot(D12 + (size_t)r * N_, w, N_, red);
    if (threadIdx.x == 0) drive[r] = -d1 + d2;
}

// Exi[r] = H31[r,:]@xi + H32[r,:]@eps + B2[r,:]@w
__global__ __launch_bounds__(256) void exi_kernel(
    const float* __restrict__ H, const float* __restrict__ B2,
    const float* __restrict__ xi, const float* __restrict__ eps,
    const float* __restrict__ w, float* __restrict__ Exi)
{
    __shared__ float red[256];
    const int r = blockIdx.x;
    const float* hrow = H + (size_t)(2 * NXI_ + r) * D3_;
    float d1 = block_dot(hrow, xi, NXI_, red);
    float d2 = block_dot(hrow + NXI_, eps, L_, red);
    float d3 = block_dot(B2 + (size_t)r * N_, w, N_, red);
    if (threadIdx.x == 0) Exi[r] = d1 + d2 + d3;
}

// u[m] = C2[m,:]@xi + D21[m,:]@eps + D22[m,:]@w
__global__ __launch_bounds__(256) void u_kernel(
    const float* __restrict__ C2, const float* __restrict__ D21,
    const float* __restrict__ D22, const float* __restrict__ xi,
    const float* __restrict__ eps, const float* __restrict__ w,
    float* __restrict__ u)
{
    __shared__ float red[256];
    const int m = blockIdx.x;
    float d1 = block_dot(C2 + (size_t)m * NXI_, xi, NXI_, red);
    float d2 = block_dot(D21 + (size_t)m * L_, eps, L_, red);
    float d3 = block_dot(D22 + (size_t)m * N_, w, N_, red);
    if (threadIdx.x == 0) u[m] = d1 + d2 + d3;
}

// ---------------------------------------------------------------------------
// Small elementwise kernels
// ---------------------------------------------------------------------------
// Mmat = XXT + Y3 - Y3.T + EPS*I ;  IpM = I + Mmat ; ImM = I - Mmat
__global__ void build_ipm_imm(const float* __restrict__ XXT,
                              const float* __restrict__ Y3,
                              float* __restrict__ IpM, float* __restrict__ ImM)
{
    int idx = blockIdx.x * blockDim.x + threadIdx.x;
    if (idx >= S_ * S_) return;
    int i = idx / S_, j = idx % S_;
    float d = (i == j) ? 1.0f : 0.0f;
    float m = XXT[idx] + Y3[(size_t)i * S_ + j] - Y3[(size_t)j * S_ + i] + d * EPS_;
    IpM[idx] = d + m;
    ImM[idx] = d - m;
}

__global__ void init_aug_inverse(const float* __restrict__ A,
                                 float* __restrict__ aug, int n)
{
    int idx = blockIdx.x * blockDim.x + threadIdx.x;
    int ncols = 2 * n;
    if (idx >= n * ncols) return;
    int i = idx / ncols, j = idx % ncols;
    aug[idx] = (j < n) ? A[(size_t)i * n + j] : ((j - n == i) ? 1.0f : 0.0f);
}

__global__ void add_diag_g(float* __restrict__ A, int n, int ld,
                           const float* __restrict__ gptr)
{
    int i = blockIdx.x * blockDim.x + threadIdx.x;
    if (i < n) A[(size_t)i * ld + i] += gptr[0];
}

__global__ void negcopy_kernel(const float* __restrict__ s,
                               float* __restrict__ d, int count)
{
    int i = blockIdx.x * blockDim.x + threadIdx.x;
    if (i < count) d[i] = -s[i];
}

__global__ void copy_kernel(const float* __restrict__ s, float* __restrict__ d,
                            int count)
{
    int i = blockIdx.x * blockDim.x + threadIdx.x;
    if (i < count) d[i] = s[i];
}

__global__ void zero_kernel(float* __restrict__ d, int count)
{
    int i = blockIdx.x * blockDim.x + threadIdx.x;
    if (i < count) d[i] = 0.0f;
}

// dst (cols x rows) = src.T, src is rows x cols
__global__ void transpose_kernel(const float* __restrict__ s,
                                 float* __restrict__ d, int rows, int cols)
{
    int idx = blockIdx.x * blockDim.x + threadIdx.x;
    if (idx >= rows * cols) return;
    int r = idx / cols, c = idx % cols;
    d[(size_t)c * rows + r] = s[idx];
}

__global__ void init_H_kernel(float* __restrict__ H)
{
    int idx = blockIdx.x * blockDim.x + threadIdx.x;
    if (idx >= D3_ * D3_) return;
    int i = idx / D3_, j = idx % D3_;
    H[idx] = (i == j) ? EPS_ : 0.0f;
}

// EAug[i][j<1024] = 0.5*(H11 + H33 + Y - Y.T)[i][j] ; EAug[i][1024] = Exi[i]
__global__ void build_eaug(const float* __restrict__ H,
                           const float* __restrict__ Y,
                           const float* __restrict__ Exi,
                           float* __restrict__ EAug)
{
    int idx = blockIdx.x * blockDim.x + threadIdx.x;
    const int ncols = NXI_ + 1;
    if (idx >= NXI_ * ncols) return;
    int i = idx / ncols, j = idx % ncols;
    if (j == NXI_) {
        EAug[idx] = Exi[i];
    } else {
        float h11 = H[(size_t)i * D3_ + j];
        float h33 = H[(size_t)(2 * NXI_ + i) * D3_ + (2 * NXI_ + j)];
        EAug[idx] = 0.5f * (h11 + h33 + Y[(size_t)i * NXI_ + j] -
                            Y[(size_t)j * NXI_ + i]);
    }
}

__global__ void extract_solution(const float* __restrict__ EAug,
                                 float* __restrict__ out)
{
    int i = blockIdx.x * blockDim.x + threadIdx.x;
    if (i < NXI_) out[i] = EAug[(size_t)i * (NXI_ + 1) + NXI_];
}

// ---------------------------------------------------------------------------
// Host orchestration
// ---------------------------------------------------------------------------
static inline dim3 gemm_grid(int Mdim, int Ndim) { return dim3(Ndim / BN, Mdim / BM); }
static inline int  ceil_div(int a, int b) { return (a + b - 1) / b; }

extern "C" void kernel_launch(void* const* d_in, const int* in_sizes, int n_in,
                              void* d_out, int out_size, void* d_ws, size_t ws_size,
                              hipStream_t stream)
{
    (void)in_sizes; (void)n_in; (void)out_size; (void)ws_size;
    const float* gamma = (const float*)d_in[0];   // (1,1) on device
    const float* X     = (const float*)d_in[1];   // 3072 x 3072
    const float* Y     = (const float*)d_in[2];   // 1024 x 1024
    const float* B2    = (const float*)d_in[3];   // 1024 x 512
    const float* C2    = (const float*)d_in[4];   // 512 x 1024
    const float* D21   = (const float*)d_in[5];   // 512 x 1024
    const float* X3    = (const float*)d_in[6];   // 512 x 512
    const float* Y3    = (const float*)d_in[7];   // 512 x 512
    const float* D12   = (const float*)d_in[8];   // 1024 x 512
    const float* w     = (const float*)d_in[9];   // 512
    const float* xi    = (const float*)d_in[10];  // 1024

    float* out_u  = (float*)d_out;        // 512
    float* out_xi = (float*)d_out + M_;   // 1024

    // Workspace layout
    float* ws = (float*)d_ws;
    size_t o = 0;
    float* XXT  = ws + o; o += (size_t)S_ * S_;
    float* IpM  = ws + o; o += (size_t)S_ * S_;
    float* ImM  = ws + o; o += (size_t)S_ * S_;
    float* AugM = ws + o; o += (size_t)S_ * 2 * S_;   // right half = (I+M)^-1
    float* D22  = ws + o; o += (size_t)M_ * N_;
    float* Rcap = ws + o; o += (size_t)N_ * N_;
    float* AugR = ws + o; o += (size_t)N_ * 2 * N_;   // right half = Rcap^-1
    float* vecR = ws + o; o += (size_t)D3_ * N_;
    float* vecQ = ws + o; o += (size_t)D3_ * M_;
    float* T2   = ws + o; o += (size_t)D3_ * N_;
    float* H    = ws + o; o += (size_t)D3_ * D3_;
    float* EAug = ws + o; o += (size_t)NXI_ * (NXI_ + 1);
    float* drv  = ws + o; o += L_;
    float* eps  = ws + o; o += L_;
    float* Exi  = ws + o; o += NXI_;

    const int T = 256;

    // 1) XXT = X3 @ X3.T
    wmma_gemm_f32<<<gemm_grid(S_, S_), 256, 0, stream>>>(
        X3, X3, XXT, S_, S_, S_, S_, S_, S_, 0, 1, 1.0f, nullptr, 0, 0.0f);

    // 2) IpM = I + Mmat, ImM = I - Mmat
    build_ipm_imm<<<ceil_div(S_ * S_, T), T, 0, stream>>>(XXT, Y3, IpM, ImM);

    // 3) (I+M)^-1 via Gauss-Jordan
    init_aug_inverse<<<ceil_div(S_ * 2 * S_, T), T, 0, stream>>>(IpM, AugM, S_);
    gauss_jordan<<<1, 1024, 0, stream>>>(AugM, S_, 2 * S_);

    // 4) D22 = g * (I-M) @ (I+M)^-1   (Zeta = M_tilde since M=N=S)
    wmma_gemm_f32<<<gemm_grid(S_, S_), 256, 0, stream>>>(
        ImM, AugM + S_, D22, S_, S_, S_, S_, 2 * S_, S_, 0, 0,
        1.0f, gamma, +1, 0.0f);

    // 5) Rcap = g*I - (1/g) D22.T @ D22
    wmma_gemm_f32<<<gemm_grid(N_, N_), 256, 0, stream>>>(
        D22, D22, Rcap, N_, N_, M_, N_, N_, N_, 1, 0, -1.0f, gamma, -1, 0.0f);
    add_diag_g<<<ceil_div(N_, T), T, 0, stream>>>(Rcap, N_, N_, gamma);

    // 6) Rcap^-1
    init_aug_inverse<<<ceil_div(N_ * 2 * N_, T), T, 0, stream>>>(Rcap, AugR, N_);
    gauss_jordan<<<1, 1024, 0, stream>>>(AugR, N_, 2 * N_);

    // 7) vec_R rows 0..1023 = C2_cap.T = -(1/g) C2.T @ D22
    wmma_gemm_f32<<<gemm_grid(NXI_, N_), 256, 0, stream>>>(
        C2, D22, vecR, NXI_, N_, M_, NXI_, N_, N_, 1, 0, -1.0f, gamma, -1, 0.0f);

    // 8) vec_R rows 1024..2047 = D21_cap.T = -(1/g) D21.T @ D22 - D12
    negcopy_kernel<<<ceil_div(L_ * N_, T), T, 0, stream>>>(
        D12, vecR + (size_t)NXI_ * N_, L_ * N_);
    wmma_gemm_f32<<<gemm_grid(L_, N_), 256, 0, stream>>>(
        D21, D22, vecR + (size_t)NXI_ * N_, L_, N_, M_, L_, N_, N_, 1, 0,
        -1.0f, gamma, -1, 1.0f);

    // 9) vec_R rows 2048..3071 = B2
    copy_kernel<<<ceil_div(NXI_ * N_, T), T, 0, stream>>>(
        B2, vecR + (size_t)2 * NXI_ * N_, NXI_ * N_);

    // 10) vec_Q = [C2.T ; D21.T ; 0]
    transpose_kernel<<<ceil_div(M_ * NXI_, T), T, 0, stream>>>(C2, vecQ, M_, NXI_);
    transpose_kernel<<<ceil_div(M_ * L_, T), T, 0, stream>>>(
        D21, vecQ + (size_t)NXI_ * M_, M_, L_);
    zero_kernel<<<ceil_div(NXI_ * M_, T), T, 0, stream>>>(
        vecQ + (size_t)2 * NXI_ * M_, NXI_ * M_);

    // 11) H = EPS*I + X.T@X
    init_H_kernel<<<ceil_div(D3_ * D3_, T), T, 0, stream>>>(H);
    wmma_gemm_f32<<<gemm_grid(D3_, D3_), 256, 0, stream>>>(
        X, X, H, D3_, D3_, D3_, D3_, D3_, D3_, 1, 0, 1.0f, nullptr, 0, 1.0f);

    // 12) T2 = vec_R @ Rcap^-1 ; H += T2 @ vec_R.T
    wmma_gemm_f32<<<gemm_grid(D3_, N_), 256, 0, stream>>>(
        vecR, AugR + N_, T2, D3_, N_, N_, N_, 2 * N_, N_, 0, 0,
        1.0f, nullptr, 0, 0.0f);
    wmma_gemm_f32<<<gemm_grid(D3_, D3_), 256, 0, stream>>>(
        T2, vecR, H, D3_, D3_, N_, N_, N_, D3_, 0, 1, 1.0f, nullptr, 0, 1.0f);

    // 13) H += (1/g) vec_Q @ vec_Q.T   (== -vec_Q @ Q @ vec_Q.T)
    wmma_gemm_f32<<<gemm_grid(D3_, D3_), 256, 0, stream>>>(
        vecQ, vecQ, H, D3_, D3_, M_, M_, M_, D3_, 0, 1, 1.0f, gamma, -1, 1.0f);

    // 14) drive = C1@xi + D12@w  (C1 = -H21)
    drive_kernel<<<L_, 256, 0, stream>>>(H, D12, xi, w, drv);

    // 15) sequential ReLU forward substitution
    seq_eps_kernel<<<1, 1024, 0, stream>>>(H, drv, eps);

    // 16) E_xi = Fm@xi + B1@eps + B2@w  (Fm=H31, B1=H32)
    exi_kernel<<<NXI_, 256, 0, stream>>>(H, B2, xi, eps, w, Exi);

    // 17) solve E @ xi_next = E_xi  (Gauss-Jordan on [E | E_xi])
    build_eaug<<<ceil_div(NXI_ * (NXI_ + 1), T), T, 0, stream>>>(H, Y, Exi, EAug);
    gauss_jordan<<<1, 1024, 0, stream>>>(EAug, NXI_, NXI_ + 1);
    extract_solution<<<ceil_div(NXI_, T), T, 0, stream>>>(EAug, out_xi);

    // 18) u = C2@xi + D21@eps + D22@w
    u_kernel<<<M_, 256, 0, stream>>>(C2, D21, D22, xi, eps, w, out_u);
}